// AVRoPE_8074538516956
// MI455X (gfx1250) — compile-verified
//
#include <hip/hip_runtime.h>

typedef float __attribute__((ext_vector_type(4))) f4;

namespace {
constexpr int kB = 4, kH = 16, kNF = 16, kP = 16, kD = 96;
constexpr int kHalf = kD / 2;                          // 48 rotation pairs per token
constexpr int kUPT  = kHalf / 4;                       // 12 work-units per token (4 pairs each)
constexpr int kVidTokens = kB * kH * kNF * kP * kP;    // 262144
constexpr int kAudTokens = kB * kH * kNF;              // 1024
constexpr int kVidUnits  = kVidTokens * kUPT;          // 3,145,728
constexpr int kAudUnits  = kAudTokens * kUPT;          // 12,288
constexpr int kTotalUnits = kVidUnits + kAudUnits;     // 3,158,016 (= 12336 * 256 exactly)
constexpr int kVidOutElems = kVidTokens * kD;          // 25,165,824
constexpr int kPP1 = kP + 1;                           // 17 (padded grid side)
}

// Memory-bound axial RoPE: y0 = x0*c - x1*s ; y1 = x1*c + x0*s,
// output channel layout [y0(0..47) | y1(0..47)].
// Streaming tensors use non-temporal (th:NT) b128 accesses; cos/sin
// (2 x 888 KB, reused 64x across B*H) use default RT loads to stay cached.
__global__ __launch_bounds__(256) void avrope_rot_kernel(
    const float* __restrict__ xv,
    const float* __restrict__ xa,
    const float* __restrict__ ct,
    const float* __restrict__ st,
    const int*  __restrict__ offp,
    float* __restrict__ out)
{
  const unsigned u = blockIdx.x * 256u + threadIdx.x;
  if (u >= (unsigned)kTotalUnits) return;
  const int offset = *offp;  // uniform scalar load

  const f4* __restrict__ xin;
  f4* o0;
  f4* o1;
  unsigned cb;  // float index into cos/sin tables

  if (u < (unsigned)kVidUnits) {
    const unsigned token = u / kUPT;           // video token id (b,h,n,py,px flattened)
    const unsigned j     = u - token * kUPT;   // which group of 4 pairs
    const unsigned sp = token & 255u;          // py*16 + px
    const unsigned n  = (token >> 8) & 15u;    // frame index
    const unsigned py = sp >> 4;
    const unsigned px = sp & 15u;
    cb = ((unsigned)(n + offset) * kPP1 + py) * kPP1 + px;
    cb = cb * kHalf + 4u * j;
    xin = (const f4*)(xv + (size_t)token * kD + 8u * j);
    float* ob = out + (size_t)token * kD;
    o0 = (f4*)(ob + 4u * j);                   // y0 half
    o1 = (f4*)(ob + kHalf + 4u * j);           // y1 half
  } else {
    const unsigned a     = u - (unsigned)kVidUnits;
    const unsigned token = a / kUPT;           // audio token id (b,h,n flattened)
    const unsigned j     = a - token * kUPT;
    const unsigned n = token & 15u;
    cb = ((unsigned)(n + offset) * kPP1 + kP) * kPP1 + kP;  // pad cell (16,16)
    cb = cb * kHalf + 4u * j;
    xin = (const f4*)(xa + (size_t)token * kD + 8u * j);
    float* ob = out + (size_t)kVidOutElems + (size_t)token * kD;
    o0 = (f4*)(ob + 4u * j);
    o1 = (f4*)(ob + kHalf + 4u * j);
  }

  // Streaming data: non-temporal b128 loads (th:NT) — no cache pollution.
  const f4 xlo = __builtin_nontemporal_load(xin);      // pairs 4j..4j+1 interleaved
  const f4 xhi = __builtin_nontemporal_load(xin + 1);  // pairs 4j+2..4j+3 interleaved
  // Tables: regular temporal loads, high reuse.
  const f4 c = *(const f4*)(ct + cb);
  const f4 s = *(const f4*)(st + cb);

  f4 y0, y1;
  y0.x = xlo.x * c.x - xlo.y * s.x;  y1.x = xlo.y * c.x + xlo.x * s.x;
  y0.y = xlo.z * c.y - xlo.w * s.y;  y1.y = xlo.w * c.y + xlo.z * s.y;
  y0.z = xhi.x * c.z - xhi.y * s.z;  y1.z = xhi.y * c.z + xhi.x * s.z;
  y0.w = xhi.z * c.w - xhi.w * s.w;  y1.w = xhi.w * c.w + xhi.z * s.w;

  __builtin_nontemporal_store(y0, o0);
  __builtin_nontemporal_store(y1, o1);
}

extern "C" void kernel_launch(void* const* d_in, const int* in_sizes, int n_in,
                              void* d_out, int out_size, void* d_ws, size_t ws_size,
                              hipStream_t stream) {
  const float* xv   = (const float*)d_in[0];  // x_video (4,16,4096,96) f32
  const float* xa   = (const float*)d_in[1];  // x_audio (4,16,16,96)   f32
  const float* ct   = (const float*)d_in[2];  // cos (16,17,17,48)      f32
  const float* st   = (const float*)d_in[3];  // sin (16,17,17,48)      f32
  const int*   offp = (const int*)d_in[4];    // offset scalar
  float* out = (float*)d_out;                 // [video_out | audio_out] f32

  const int blocks = (kTotalUnits + 255) / 256;  // 12336
  avrope_rot_kernel<<<blocks, 256, 0, stream>>>(xv, xa, ct, st, offp, out);
}